// DynamicMaskHead_87969520157398
// MI455X (gfx1250) — compile-verified
//
#include <hip/hip_runtime.h>

// DynamicMaskHead (CondInst-style) for MI455X / gfx1250, wave32.
//
// Two instances packed per 16x16 WMMA tile (they share the feature map, so
// they share the B operand in layer 1):
//   layer1: A(16x8)  = [W0_a ; W0_b],          B(8x16) = features  (2 WMMAs)
//   layer2: A(16x16) = column-permuted blockdiag(W1_a, W1_b),
//           B        = layer-1 D registers FED DIRECTLY (4 WMMAs, no shuffles)
// using V_WMMA_F32_16X16X4_F32 (exact f32 path, matches reference numerics).
//
// Layer-2 trick: a D-layout vgpr r holds (instA ch r, pixel L) in lanes 0-15
// and (instB ch r, pixel L-16) in lanes 16-31 — exactly the pixel striping the
// B operand expects. Reinterpreting pair (d[2q], d[2q+1]) as B chunk q makes
// the hardware read K=4q,4q+1 as instA channels 2q,2q+1 and K=4q+2,4q+3 as
// instB channels 2q,2q+1; A2's columns are permuted to match. This removes
// all cross-lane data movement between the two layers.
//
// Sigmoid uses v_exp_f32 + v_rcp_f32 (1-ulp) instead of IEEE division: the
// div_scale/NR-refine sequence was ~10 extra VALU ops per tile.

typedef float v2f __attribute__((ext_vector_type(2)));
typedef float v8f __attribute__((ext_vector_type(8)));

#define NPAR 153
#define HW   25600   // 160*160
#define NTILE (HW / 16)

static __device__ __forceinline__ v8f wmma4(v2f a, v2f b, v8f c) {
  // D(16x16,f32) += A(16x4,f32) * B(4x16,f32)
  return __builtin_amdgcn_wmma_f32_16x16x4_f32(
      /*neg_a=*/false, a, /*neg_b=*/false, b,
      /*c_mod=*/(short)0, c, /*reuse_a=*/false, /*reuse_b=*/false);
}

__global__ __launch_bounds__(256) void DynamicMaskHead_gfx1250_kernel(
    const float* __restrict__ feats,   // [8][160][160], shared by all instances
    const float* __restrict__ params,  // [n_inst][153]
    float* __restrict__ out,           // [n_inst][160][160]
    int n_inst) {
  __shared__ float s_p[2][NPAR];  // tanh(params) for the block's two instances

  const int iA = 2 * blockIdx.x;
  const int iB = iA + 1;
  const bool hasB = (iB < n_inst);

  const int tid = threadIdx.x;
  // Cooperative tanh of both instances' params into LDS.
  for (int i = tid; i < 2 * NPAR; i += 256) {
    const int which = i / NPAR;
    const int j = i - which * NPAR;
    const int g = (which && hasB) ? iB : iA;  // duplicate A if odd tail
    s_p[which][j] = tanhf(params[g * NPAR + j]);
  }
  __syncthreads();

  const int lane = tid & 31;
  const int wave = tid >> 5;
  const bool lo = (lane < 16);   // half-wave select
  const int m = lane & 15;       // WMMA A-matrix row M for this lane
  const int mr = m & 7;          // row within the instance's 8x8 weight
  const float* __restrict__ pm = s_p[(m < 8) ? 0 : 1];  // params for A row M
  const float* __restrict__ ps = s_p[lo ? 0 : 1];       // params for D/B half
  const int kb = lo ? 0 : 2;     // A 16x4 layout: lanes16-31 hold K+2

  // ---- Layer-1 A operand: A[M][K] = W0_inst(M)[mr][K], K split 0..3 / 4..7
  v2f a1lo, a1hi;
  a1lo.x = pm[mr * 8 + kb + 0];
  a1lo.y = pm[mr * 8 + kb + 1];
  a1hi.x = pm[mr * 8 + kb + 4];
  a1hi.y = pm[mr * 8 + kb + 5];

  // ---- Layer-2 A operand (column-permuted block-diagonal):
  // chunk q, lane-lo slots are K=4q,4q+1  -> instA channels 2q,2q+1 (rows M<8)
  //          lane-hi slots are K=4q+2,4q+3-> instB channels 2q,2q+1 (rows M>=8)
  // valid iff this lane's A row instance matches the K slot's instance:
  const bool a2valid = ((m < 8) == lo);
  v2f a2[4];
#pragma unroll
  for (int q = 0; q < 4; ++q) {
    a2[q].x = a2valid ? pm[64 + mr * 8 + 2 * q + 0] : 0.0f;
    a2[q].y = a2valid ? pm[64 + mr * 8 + 2 * q + 1] : 0.0f;
  }

  // ---- Per-lane biases / final weights (C/D layout: vgpr r, lanes0-15 ->
  // instA channel r, lanes16-31 -> instB channel r).
  float b0v[8], b1v[8], w2v[8];
#pragma unroll
  for (int r = 0; r < 8; ++r) {
    w2v[r] = ps[128 + r];
    b0v[r] = ps[136 + r];
    b1v[r] = ps[144 + r];
  }
  const float b2v = ps[152];

  const int pixin = lane & 15;  // pixel within 16-wide tile (both halves)
  const int cb = lo ? 0 : 2;    // B 4x16 layout: lanes16-31 hold K+2
  // If no instance B, lanes 16-31 duplicate instance A (identical values to
  // identical addresses -> benign), so the store is unconditional.
  const int outInst = (!lo && hasB) ? iB : iA;

  // 1600 tiles of 16 pixels; 8 waves -> 200 uniform iterations each, so EXEC
  // is all-ones at every WMMA as the ISA requires.
  for (int t = wave; t < NTILE; t += 8) {
    const int p = t * 16 + pixin;

    // ---- Layer-1 B operand: features, channels (cb, cb+1, cb+4, cb+5);
    // each load is two 64B fully-coalesced half-wave segments (L2 resident).
    v2f bl, bh;
    bl.x = feats[(cb + 0) * HW + p];
    bl.y = feats[(cb + 1) * HW + p];
    bh.x = feats[(cb + 4) * HW + p];
    bh.y = feats[(cb + 5) * HW + p];
    // Prefetch this wave's next tile (speculative; dropped if OOB).
    __builtin_prefetch(&feats[(cb + 0) * HW + p + 128], 0, 1);

    // ---- Layer 1: y0 = [W0_a;W0_b] * x   (K=8 via 2 chained WMMAs)
    v8f acc = {};
    acc = wmma4(a1lo, bl, acc);
    acc = wmma4(a1hi, bh, acc);

    // Bias + ReLU in the D layout; results feed layer 2 B operands directly.
    float d[8];
#pragma unroll
    for (int r = 0; r < 8; ++r) d[r] = fmaxf(acc[r] + b0v[r], 0.0f);

    // ---- Layer 2: y1 = A2 * relu(y0+b0), B chunks are (d[2q], d[2q+1])
    v8f acc2 = {};
    acc2 = wmma4(a2[0], (v2f){d[0], d[1]}, acc2);
    acc2 = wmma4(a2[1], (v2f){d[2], d[3]}, acc2);
    acc2 = wmma4(a2[2], (v2f){d[4], d[5]}, acc2);
    acc2 = wmma4(a2[3], (v2f){d[6], d[7]}, acc2);

    // ---- Layer 3: sigmoid(w2 . relu(y1+b1) + b2), per lane (16 px x 2 inst)
    float z = b2v;
#pragma unroll
    for (int r = 0; r < 8; ++r)
      z = fmaf(w2v[r], fmaxf(acc2[r] + b1v[r], 0.0f), z);
    // Fast sigmoid: v_exp_f32 + v_rcp_f32 (avoids IEEE div_scale/NR sequence).
    const float o = __builtin_amdgcn_rcpf(1.0f + __expf(-z));

    // lanes0-15 -> instance A, lanes16-31 -> instance B; 64B segments/half.
    out[outInst * HW + p] = o;
  }
}

extern "C" void kernel_launch(void* const* d_in, const int* in_sizes, int n_in,
                              void* d_out, int out_size, void* d_ws, size_t ws_size,
                              hipStream_t stream) {
  const float* feats  = (const float*)d_in[0];  // mask_feats  [1,8,160,160]
  const float* params = (const float*)d_in[1];  // mask_head_params [n,153]
  // d_in[2] (det_bbox) and d_in[3] (fpn_levels) are unused by the reference.
  float* out = (float*)d_out;

  const int n_inst = in_sizes[1] / NPAR;
  const int blocks = (n_inst + 1) / 2;  // 2 instances per workgroup
  DynamicMaskHead_gfx1250_kernel<<<blocks, 256, 0, stream>>>(feats, params, out,
                                                             n_inst);
}